// OptimizedS4DKernel_8804682957529
// MI455X (gfx1250) — compile-verified
//
#include <hip/hip_runtime.h>

typedef float v2f __attribute__((ext_vector_type(2)));
typedef float v8f __attribute__((ext_vector_type(8)));

#define D_MODEL 512
#define NSTATE  32
#define SEQ_LEN 8192
#define WAVES_PER_BLOCK 4
#define L_PER_WAVE (SEQ_LEN / WAVES_PER_BLOCK)   // 2048
#define TILES_PER_WAVE (L_PER_WAVE / 256)        // 8 macro tiles of 256 l each

// Branch-free angle for sin/cos of y*l: reduce to [-pi, pi] in fp64
// (setup-only; avoids divergent libm Payne-Hanek slow paths).
__device__ __forceinline__ float reduced_angle(float y, float l) {
    const double TWO_PI     = 6.28318530717958647692528676655900577;
    const double INV_TWO_PI = 0.15915494309189533576888376337251436;
    double ang = (double)y * (double)l;
    double k   = __builtin_rint(ang * INV_TWO_PI);   // v_rndne_f64
    double red = __builtin_fma(-TWO_PI, k, ang);     // v_fma_f64
    return (float)red;
}

// S4D kernel generation restructured as per-d GEMM:
//   K[d, l0 + 16m + j] = sum_{k<64} A[m,k] * B[k,j]
//   A[m, n]     = Re(2*Cc_n * exp(dtA_n*(l0+16m)))   (n<32)
//   A[m, n+32]  = Im(2*Cc_n * exp(dtA_n*(l0+16m)))
//   B[n, j]     =  Re(exp(dtA_n * j))
//   B[n+32, j]  = -Im(exp(dtA_n * j))
// computed with 16 chained V_WMMA_F32_16X16X4_F32 per 16x16 output tile.
__global__ __launch_bounds__(32 * WAVES_PER_BLOCK) void s4d_wmma_kernel(
    const float* __restrict__ log_dt,
    const float* __restrict__ log_A_real,
    const float* __restrict__ A_imag,
    const float* __restrict__ C,
    float* __restrict__ K)
{
    const int d    = blockIdx.x;
    const int wave = threadIdx.x >> 5;
    const int lane = threadIdx.x & 31;
    const int h    = lane >> 4;     // half-wave selector (K-pair within chunk)
    const int MN   = lane & 15;     // A-matrix row M == B-matrix column N

    const float dt = __expf(log_dt[d]);

    float sr[16], si[16];        // A-row state: 2*Cc_n * exp(dtA_n * l_row)
    float w256r[16], w256i[16];  // per-macro-tile advance: exp(dtA_n * 256)
    float Br[16], Bi[16];        // B values at column N:  Re(w^N), -Im(w^N)

    const int   wave_l0 = wave * L_PER_WAVE;
    const float lrow    = (float)(wave_l0 + 16 * MN);
    const float ncol    = (float)MN;

    // This lane's 16 distinct state indices n_q = 4*(q>>1) + 2h + (q&1):
    // chunk t<8  uses A cols {4t+2h, 4t+2h+1}       -> sr[2t], sr[2t+1]
    // chunk t>=8 uses A cols {4t+2h, ...} - 32      -> si[2(t-8)], si[2(t-8)+1]
    #pragma unroll
    for (int q = 0; q < 16; ++q) {
        const int n = 4 * (q >> 1) + 2 * h + (q & 1);
        const float lar = log_A_real[d * NSTATE + n];
        const float aim = A_imag[d * NSTATE + n];
        const float Ar  = -__expf(lar);
        const float x   = dt * Ar;    // Re(dtA), always <= 0
        const float y   = dt * aim;   // Im(dtA), in [0, ~0.32]

        // Discretized C: Cc = C * (exp(dtA) - 1) / (A + 1e-8), folded *2
        const float em = __expf(x);
        float sy, cy; __sincosf(y, &sy, &cy);           // small angle, native
        const float er = em * cy - 1.0f;
        const float ei = em * sy;
        const float dr = Ar + 1e-8f;
        const float di = aim;
        const float inv_den = 2.0f / (dr * dr + di * di);   // fold the *2 here
        const float tr = (er * dr + ei * di) * inv_den;
        const float ti = (ei * dr - er * di) * inv_den;
        const float Cr = C[(d * NSTATE + n) * 2 + 0];
        const float Ci = C[(d * NSTATE + n) * 2 + 1];
        const float ccr = Cr * tr - Ci * ti;
        const float cci = Cr * ti + Ci * tr;

        // s init = 2Cc * exp(dtA * lrow); fp64-reduced angle, native sincos
        const float mag0 = __expf(x * lrow);
        const float a0   = reduced_angle(y, lrow);
        float s0, c0; __sincosf(a0, &s0, &c0);
        const float E0r = mag0 * c0, E0i = mag0 * s0;
        sr[q] = ccr * E0r - cci * E0i;
        si[q] = ccr * E0i + cci * E0r;

        // Exact macro-tile advance factor w^256 (fp64-reduced angle)
        const float mag2 = __expf(x * 256.0f);
        const float a2   = reduced_angle(y, 256.0f);
        float s2, c2; __sincosf(a2, &s2, &c2);
        w256r[q] = mag2 * c2;
        w256i[q] = mag2 * s2;

        // B column values: w^N, angle <= ~4.8 rad -> native sincos directly
        const float magb = __expf(x * ncol);
        const float angb = y * ncol;
        float sb, cb; __sincosf(angb, &sb, &cb);
        Br[q] =  magb * cb;
        Bi[q] = -magb * sb;
    }

    float* out_base = K + (size_t)d * SEQ_LEN;

    #pragma unroll
    for (int tile = 0; tile < TILES_PER_WAVE; ++tile) {
        const int l0 = wave_l0 + tile * 256;
        v8f c = {0.f, 0.f, 0.f, 0.f, 0.f, 0.f, 0.f, 0.f};

        // Real-part contraction: chunks t = 0..7 (A cols 0..31)
        #pragma unroll
        for (int t = 0; t < 8; ++t) {
            v2f a, b;
            a[0] = sr[2 * t]; a[1] = sr[2 * t + 1];
            b[0] = Br[2 * t]; b[1] = Br[2 * t + 1];
            c = __builtin_amdgcn_wmma_f32_16x16x4_f32(
                    false, a, false, b, (short)0, c, false, false);
        }
        // Imag-part contraction: chunks t = 8..15 (A cols 32..63)
        #pragma unroll
        for (int t = 0; t < 8; ++t) {
            v2f a, b;
            a[0] = si[2 * t]; a[1] = si[2 * t + 1];
            b[0] = Bi[2 * t]; b[1] = Bi[2 * t + 1];
            c = __builtin_amdgcn_wmma_f32_16x16x4_f32(
                    false, a, false, b, (short)0, c, false, false);
        }

        // D-tile store: lane holds D[v + 8h][MN] in c[v]; l = l0 + 16*M + N.
        // Each half-wave writes 16 consecutive floats (64B) per v.
        #pragma unroll
        for (int v = 0; v < 8; ++v) {
            out_base[l0 + 16 * (v + 8 * h) + MN] = c[v];
        }

        // Advance A-row state to next macro tile: s *= w^256 (complex)
        if (tile != TILES_PER_WAVE - 1) {
            #pragma unroll
            for (int q = 0; q < 16; ++q) {
                const float nr = sr[q] * w256r[q] - si[q] * w256i[q];
                const float ni = sr[q] * w256i[q] + si[q] * w256r[q];
                sr[q] = nr; si[q] = ni;
            }
        }
    }
}

extern "C" void kernel_launch(void* const* d_in, const int* in_sizes, int n_in,
                              void* d_out, int out_size, void* d_ws, size_t ws_size,
                              hipStream_t stream) {
    const float* log_dt     = (const float*)d_in[0];
    const float* log_A_real = (const float*)d_in[1];
    const float* A_imag     = (const float*)d_in[2];
    const float* C          = (const float*)d_in[3];
    // d_in[4] is L (scalar) -- compile-time constant SEQ_LEN here.
    float* out = (float*)d_out;

    dim3 grid(D_MODEL);                 // one block per d
    dim3 block(32 * WAVES_PER_BLOCK);   // 4 waves, each owns 2048 l-values
    hipLaunchKernelGGL(s4d_wmma_kernel, grid, block, 0, stream,
                       log_dt, log_A_real, A_imag, C, out);
}